// AdaptiveSoftmax_54451595379257
// MI455X (gfx1250) — compile-verified
//
#include <hip/hip_runtime.h>
#include <math.h>

// AdaptiveSoftmax on gfx1250 using V_WMMA_F32_16X16X4_F32 (exact fp32 math).
// v3: 32 rows x 64 cols per wave (6 loads : 8 WMMAs), software-pipelined
// k-loop pinned with sched_barrier(0) so loads for step k+4 are issued and
// in flight while step k's WMMAs execute (true double buffering).

#define CUT0 10000
#define CUT1 30000
#define DIM  1024
#define NROWS 4096
#define HEADM 10002
#define M1 20000
#define M2 20257
#define D1 256
#define D2 64
#define ROWS_B 32            // rows per block (2 row-tiles of 16)
#define CHUNK 512            // columns per block (8 waves x 4 tiles x 16)
#define NCH_H 20             // ceil(10002/512)
#define NCH_1 40             // ceil(20000/512)
#define NCH_2 40             // ceil(20257/512)

typedef float v2f __attribute__((ext_vector_type(2)));
typedef float v8f __attribute__((ext_vector_type(8)));

__device__ __forceinline__ v8f wmma4(v2f a, v2f b, v8f c) {
  // (neg_a, A, neg_b, B, c_mod, C, reuse_a, reuse_b)
  return __builtin_amdgcn_wmma_f32_16x16x4_f32(false, a, false, b, (short)0, c,
                                               false, false);
}

// ---------------------------------------------------------------------------
// Core: one wave computes a 32x64 tile of C = X @ W^T with a pipelined k-loop.
// Pointers are per-lane fragment bases (already offset by 2*half).
// Order enforced per step: [6 loads for k+4] | fence | [8 WMMAs for k] | fence
// ---------------------------------------------------------------------------
__device__ __forceinline__ void gemm_wave32(const float* xp0, const float* xp1,
                                            const float* wp0, const float* wp1,
                                            const float* wp2, const float* wp3,
                                            int K, v8f acc[2][4]) {
  v2f a0 = *(const v2f*)(xp0);
  v2f a1 = *(const v2f*)(xp1);
  v2f b0 = *(const v2f*)(wp0);
  v2f b1 = *(const v2f*)(wp1);
  v2f b2 = *(const v2f*)(wp2);
  v2f b3 = *(const v2f*)(wp3);
  for (int k = 4; k < K; k += 4) {
    v2f a0n = *(const v2f*)(xp0 + k);
    v2f a1n = *(const v2f*)(xp1 + k);
    v2f b0n = *(const v2f*)(wp0 + k);
    v2f b1n = *(const v2f*)(wp1 + k);
    v2f b2n = *(const v2f*)(wp2 + k);
    v2f b3n = *(const v2f*)(wp3 + k);
    __builtin_amdgcn_sched_barrier(0);   // keep k+4 loads above k's WMMAs
    acc[0][0] = wmma4(a0, b0, acc[0][0]);
    acc[0][1] = wmma4(a0, b1, acc[0][1]);
    acc[0][2] = wmma4(a0, b2, acc[0][2]);
    acc[0][3] = wmma4(a0, b3, acc[0][3]);
    acc[1][0] = wmma4(a1, b0, acc[1][0]);
    acc[1][1] = wmma4(a1, b1, acc[1][1]);
    acc[1][2] = wmma4(a1, b2, acc[1][2]);
    acc[1][3] = wmma4(a1, b3, acc[1][3]);
    __builtin_amdgcn_sched_barrier(0);   // keep next loads below these WMMAs
    a0 = a0n; a1 = a1n; b0 = b0n; b1 = b1n; b2 = b2n; b3 = b3n;
  }
  acc[0][0] = wmma4(a0, b0, acc[0][0]);
  acc[0][1] = wmma4(a0, b1, acc[0][1]);
  acc[0][2] = wmma4(a0, b2, acc[0][2]);
  acc[0][3] = wmma4(a0, b3, acc[0][3]);
  acc[1][0] = wmma4(a1, b0, acc[1][0]);
  acc[1][1] = wmma4(a1, b1, acc[1][1]);
  acc[1][2] = wmma4(a1, b2, acc[1][2]);
  acc[1][3] = wmma4(a1, b3, acc[1][3]);
}

// ---------------------------------------------------------------------------
// C[N x M] = X[N x K] @ W[M x K]^T, stored dense. M multiple of 64, N mult 32.
// Block: 256 threads (8 waves). Each wave: 32 rows x 64 cols (8 WMMA tiles).
// ---------------------------------------------------------------------------
__global__ void gemm_store(const float* __restrict__ X, const float* __restrict__ W,
                           float* __restrict__ C, int K, int M) {
  const int wave = threadIdx.x >> 5;
  const int lane = threadIdx.x & 31;
  const int half = lane >> 4;     // 0: lanes 0-15, 1: lanes 16-31
  const int r    = lane & 15;
  const int row0 = blockIdx.y * ROWS_B;
  const int col0 = blockIdx.x * CHUNK + wave * 64;
  if (col0 >= M) return;

  const float* xp0 = X + (size_t)(row0 + r) * K + 2 * half;
  const float* xp1 = X + (size_t)(row0 + 16 + r) * K + 2 * half;
  const float* wp0 = W + (size_t)(col0 + r) * K + 2 * half;
  const float* wp1 = W + (size_t)(col0 + 16 + r) * K + 2 * half;
  const float* wp2 = W + (size_t)(col0 + 32 + r) * K + 2 * half;
  const float* wp3 = W + (size_t)(col0 + 48 + r) * K + 2 * half;

  v8f acc[2][4];
#pragma unroll
  for (int rt = 0; rt < 2; ++rt)
#pragma unroll
    for (int t = 0; t < 4; ++t)
      acc[rt][t] = (v8f){0.f,0.f,0.f,0.f,0.f,0.f,0.f,0.f};
  gemm_wave32(xp0, xp1, wp0, wp1, wp2, wp3, K, acc);

  // C layout: vgpr v -> row = v + 8*half, col = r (per 16-col tile)
#pragma unroll
  for (int rt = 0; rt < 2; ++rt) {
#pragma unroll
    for (int t = 0; t < 4; ++t) {
#pragma unroll
      for (int v = 0; v < 8; ++v) {
        int rr = row0 + 16 * rt + v + 8 * half;
        C[(size_t)rr * M + col0 + 16 * t + r] = acc[rt][t][v];
      }
    }
  }
}

// ---------------------------------------------------------------------------
// Same GEMM core, reduces exp(logit) over this block's 512-column chunk into
// partial[row * nchunks + chunk]. Ragged M: clamped loads + per-element mask.
// ---------------------------------------------------------------------------
__global__ void gemm_expsum(const float* __restrict__ X, const float* __restrict__ W,
                            float* __restrict__ partial, int K, int M, int nchunks) {
  __shared__ float lsum[8][ROWS_B];
  const int wave = threadIdx.x >> 5;
  const int lane = threadIdx.x & 31;
  const int half = lane >> 4;
  const int r    = lane & 15;
  const int row0 = blockIdx.y * ROWS_B;
  const int col0 = blockIdx.x * CHUNK + wave * 64;
  const bool active = (col0 < M);

  v8f acc[2][4];
#pragma unroll
  for (int rt = 0; rt < 2; ++rt)
#pragma unroll
    for (int t = 0; t < 4; ++t)
      acc[rt][t] = (v8f){0.f,0.f,0.f,0.f,0.f,0.f,0.f,0.f};
  if (active) {
    const float* xp0 = X + (size_t)(row0 + r) * K + 2 * half;
    const float* xp1 = X + (size_t)(row0 + 16 + r) * K + 2 * half;
    const float* wp0 = W + (size_t)min(col0 + r,      M - 1) * K + 2 * half;
    const float* wp1 = W + (size_t)min(col0 + 16 + r, M - 1) * K + 2 * half;
    const float* wp2 = W + (size_t)min(col0 + 32 + r, M - 1) * K + 2 * half;
    const float* wp3 = W + (size_t)min(col0 + 48 + r, M - 1) * K + 2 * half;
    gemm_wave32(xp0, xp1, wp0, wp1, wp2, wp3, K, acc);
  }
  // Per-row exp-sum: a tile's 16 columns live in one 16-lane half; xor-reduce
  // within halves (masks < 16 never cross the half boundary).
#pragma unroll
  for (int rt = 0; rt < 2; ++rt) {
#pragma unroll
    for (int v = 0; v < 8; ++v) {
      float e = 0.f;
      if (active) {
#pragma unroll
        for (int t = 0; t < 4; ++t)
          if (col0 + 16 * t + r < M) e += expf(acc[rt][t][v]);
      }
      e += __shfl_xor(e, 1, 32);
      e += __shfl_xor(e, 2, 32);
      e += __shfl_xor(e, 4, 32);
      e += __shfl_xor(e, 8, 32);
      if (r == 0) lsum[wave][16 * rt + v + 8 * half] = e;  // lanes 0 and 16
    }
  }
  __syncthreads();
  if (threadIdx.x < ROWS_B) {
    float t = 0.f;
#pragma unroll
    for (int w = 0; w < 8; ++w) t += lsum[w][threadIdx.x];
    partial[(size_t)(row0 + threadIdx.x) * nchunks + blockIdx.x] = t;
  }
}

// ---------------------------------------------------------------------------
// One wave per row: the three gathered logits (head + both tails, clamped).
// ---------------------------------------------------------------------------
__global__ void gather_dots(const float* __restrict__ x, const int* __restrict__ target,
                            const float* __restrict__ Wh, const float* __restrict__ Wo1,
                            const float* __restrict__ Wo2, const float* __restrict__ h1,
                            const float* __restrict__ h2,
                            float* __restrict__ dh, float* __restrict__ d1,
                            float* __restrict__ d2) {
  const int wave = threadIdx.x >> 5;
  const int lane = threadIdx.x & 31;
  const int row  = blockIdx.x * 8 + wave;
  const int t = target[row];
  const int g = (t < CUT0) ? t : ((t < CUT1) ? CUT0 : CUT0 + 1);

  float s0 = 0.f, s1 = 0.f, s2 = 0.f;
  {
    const float* xr = x + (size_t)row * DIM;
    const float* wr = Wh + (size_t)g * DIM;
    for (int k = lane; k < DIM; k += 32) s0 += xr[k] * wr[k];
  }
  {
    const int rel = min(max(t - CUT0, 0), M1 - 1);
    const float* hr = h1 + (size_t)row * D1;
    const float* wr = Wo1 + (size_t)rel * D1;
    for (int k = lane; k < D1; k += 32) s1 += hr[k] * wr[k];
  }
  {
    const int rel = min(max(t - CUT1, 0), M2 - 1);
    const float* hr = h2 + (size_t)row * D2;
    const float* wr = Wo2 + (size_t)rel * D2;
    for (int k = lane; k < D2; k += 32) s2 += hr[k] * wr[k];
  }
#pragma unroll
  for (int m = 16; m; m >>= 1) {
    s0 += __shfl_xor(s0, m, 32);
    s1 += __shfl_xor(s1, m, 32);
    s2 += __shfl_xor(s2, m, 32);
  }
  if (lane == 0) { dh[row] = s0; d1[row] = s1; d2[row] = s2; }
}

// ---------------------------------------------------------------------------
// out[i] = dot_g - log(sum exp)  (+ masked tail term)
// ---------------------------------------------------------------------------
__global__ void finalize(const int* __restrict__ target,
                         const float* __restrict__ ph, const float* __restrict__ p1,
                         const float* __restrict__ p2, const float* __restrict__ dh,
                         const float* __restrict__ d1, const float* __restrict__ d2,
                         float* __restrict__ out) {
  const int i = blockIdx.x * blockDim.x + threadIdx.x;
  if (i >= NROWS) return;
  float sh = 0.f;
  for (int c = 0; c < NCH_H; ++c) sh += ph[(size_t)i * NCH_H + c];
  float o = dh[i] - logf(sh);
  const int t = target[i];
  if (t >= CUT0 && t < CUT1) {
    float s = 0.f;
    for (int c = 0; c < NCH_1; ++c) s += p1[(size_t)i * NCH_1 + c];
    o += d1[i] - logf(s);
  } else if (t >= CUT1) {
    float s = 0.f;
    for (int c = 0; c < NCH_2; ++c) s += p2[(size_t)i * NCH_2 + c];
    o += d2[i] - logf(s);
  }
  out[i] = o;
}

__global__ void loss_reduce(const float* __restrict__ out, float* __restrict__ loss) {
  __shared__ float sm[256];
  float s = 0.f;
  for (int i = threadIdx.x; i < NROWS; i += 256) s += out[i];
  sm[threadIdx.x] = s;
  __syncthreads();
  for (int w = 128; w; w >>= 1) {
    if (threadIdx.x < w) sm[threadIdx.x] += sm[threadIdx.x + w];
    __syncthreads();
  }
  if (threadIdx.x == 0) *loss = -sm[0] / (float)NROWS;
}

extern "C" void kernel_launch(void* const* d_in, const int* in_sizes, int n_in,
                              void* d_out, int out_size, void* d_ws, size_t ws_size,
                              hipStream_t stream) {
  (void)in_sizes; (void)n_in; (void)out_size; (void)ws_size;
  const float* x    = (const float*)d_in[0];
  const int*   targ = (const int*)d_in[1];
  const float* Wh   = (const float*)d_in[2];
  const float* Wp1  = (const float*)d_in[3];
  const float* Wo1  = (const float*)d_in[4];
  const float* Wp2  = (const float*)d_in[5];
  const float* Wo2  = (const float*)d_in[6];
  float* out = (float*)d_out;

  float* ws = (float*)d_ws;
  float* h1 = ws;                         // 4096*256
  float* h2 = h1 + (size_t)NROWS * D1;    // 4096*64
  float* ph = h2 + (size_t)NROWS * D2;    // 4096*NCH_H
  float* p1 = ph + (size_t)NROWS * NCH_H; // 4096*NCH_1
  float* p2 = p1 + (size_t)NROWS * NCH_1; // 4096*NCH_2
  float* dh = p2 + (size_t)NROWS * NCH_2; // 4096
  float* d1 = dh + NROWS;                 // 4096
  float* d2 = d1 + NROWS;                 // 4096

  const dim3 blk(256);
  const int rowBlocks = NROWS / ROWS_B;   // 128
  // Tail hidden projections: h1 = x @ Wp1^T, h2 = x @ Wp2^T
  gemm_store<<<dim3(1, rowBlocks), blk, 0, stream>>>(x, Wp1, h1, DIM, D1);
  gemm_store<<<dim3(1, rowBlocks), blk, 0, stream>>>(x, Wp2, h2, DIM, D2);
  // Exp-sum partials for all three softmaxes
  gemm_expsum<<<dim3(NCH_H, rowBlocks), blk, 0, stream>>>(x,  Wh,  ph, DIM, HEADM, NCH_H);
  gemm_expsum<<<dim3(NCH_1, rowBlocks), blk, 0, stream>>>(h1, Wo1, p1, D1,  M1,    NCH_1);
  gemm_expsum<<<dim3(NCH_2, rowBlocks), blk, 0, stream>>>(h2, Wo2, p2, D2,  M2,    NCH_2);
  // Gathered logits (needs h1/h2)
  gather_dots<<<dim3(NROWS / 8), blk, 0, stream>>>(x, targ, Wh, Wo1, Wo2, h1, h2, dh, d1, d2);
  // Combine + loss
  finalize<<<dim3(NROWS / 256), blk, 0, stream>>>(targ, ph, p1, p2, dh, d1, d2, out);
  loss_reduce<<<1, 256, 0, stream>>>(out, out + NROWS);
}